// GAT_4861902979553
// MI455X (gfx1250) — compile-verified
//
#include <hip/hip_runtime.h>
#include <hip/hip_bf16.h>
#include <math.h>

// ---------------------------------------------------------------------------
// GAT 2-layer pipeline for gfx1250 (MI455X).
//   GEMMs via V_WMMA_F32_16X16X4_F32 (exact fp32 matrix path, wave32).
//   Edge softmax: ordered-uint atomicMax -> exp/atomicAdd -> normalized scatter.
// ---------------------------------------------------------------------------

typedef float v2f __attribute__((ext_vector_type(2)));
typedef float v8f __attribute__((ext_vector_type(8)));

static __device__ __forceinline__ unsigned fenc(float f) {
    unsigned u = __float_as_uint(f);
    return (u & 0x80000000u) ? ~u : (u | 0x80000000u);
}
static __device__ __forceinline__ float fdec(unsigned u) {
    u = (u & 0x80000000u) ? (u & 0x7fffffffu) : ~u;
    return __uint_as_float(u);
}

// ---------------------------------------------------------------------------
// WMMA fp32 GEMM:  Hout[M,Nout] = A[M,K] * W[Nout,K]^T   (h = x @ W.T)
// One wave computes one 16x16 output tile; K stepped by 4 per WMMA.
// A-frag: lane l(0..15) half h(0/1): float2 = A[m0+l][k+2h .. k+2h+1]
// B-frag: float2 = W[n0+l][k+2h .. k+2h+1]  (B[k][n] = W[n][k])
// C/D: VGPR j -> row m0 + j + 8*half, col n0 + (lane&15)
// ---------------------------------------------------------------------------
__global__ void gat_gemm_wmma_f32(const float* __restrict__ A,
                                  const float* __restrict__ W,
                                  float* __restrict__ Hout,
                                  int M, int K, int Nout)
{
    const int lane      = threadIdx.x & 31;
    const int waveInBlk = threadIdx.x >> 5;
    const int tilesN    = Nout >> 4;
    const int numTiles  = ((M + 15) >> 4) * tilesN;
    const int tile      = blockIdx.x * (blockDim.x >> 5) + waveInBlk;
    if (tile >= numTiles) return;              // wave-uniform: EXEC stays full

    const int mt   = tile / tilesN;
    const int nt   = tile - mt * tilesN;
    const int m0   = mt << 4;
    const int n0   = nt << 4;
    const int half = lane >> 4;
    const int l    = lane & 15;

    int rowA = m0 + l; if (rowA >= M) rowA = M - 1;   // clamp tail reads
    const float* aPtr = A + (size_t)rowA * K + 2 * half;
    const float* bPtr = W + (size_t)(n0 + l) * K + 2 * half;

    v8f acc = {0.f, 0.f, 0.f, 0.f, 0.f, 0.f, 0.f, 0.f};
    for (int k = 0; k < K; k += 4) {
        v2f a = *(const v2f*)(aPtr + k);
        v2f b = *(const v2f*)(bPtr + k);
        acc = __builtin_amdgcn_wmma_f32_16x16x4_f32(
            /*neg_a=*/false, a, /*neg_b=*/false, b,
            /*c_mod=*/(short)0, acc, /*reuse_a=*/false, /*reuse_b=*/false);
    }

#pragma unroll
    for (int j = 0; j < 8; ++j) {
        int row = m0 + j + 8 * half;
        if (row < M) Hout[(size_t)row * Nout + n0 + l] = acc[j];
    }
}

// ---------------------------------------------------------------------------
// Per-node attention dots: a_src[n,h] = <H[n,h,:], att_src[h,:]>, same a_dst.
// One wave per (node, head); wave-wide shuffle reduction (wave32).
// ---------------------------------------------------------------------------
__global__ void gat_attn_dots(const float* __restrict__ Hf,
                              const float* __restrict__ att_src,
                              const float* __restrict__ att_dst,
                              float* __restrict__ a_src,
                              float* __restrict__ a_dst,
                              int N, int H, int C)
{
    const int wave = (int)((blockIdx.x * blockDim.x + threadIdx.x) >> 5);
    const int lane = threadIdx.x & 31;
    const int total = N * H;
    if (wave >= total) return;
    const int n  = wave / H;
    const int hd = wave - n * H;

    const float* hp = Hf + (size_t)n * H * C + (size_t)hd * C;
    const float* as = att_src + hd * C;
    const float* ad = att_dst + hd * C;

    float s = 0.f, d = 0.f;
    for (int c = lane; c < C; c += 32) {
        float v = hp[c];
        s += v * as[c];
        d += v * ad[c];
    }
#pragma unroll
    for (int off = 16; off > 0; off >>= 1) {
        s += __shfl_xor(s, off, 32);
        d += __shfl_xor(d, off, 32);
    }
    if (lane == 0) { a_src[wave] = s; a_dst[wave] = d; }
}

// ---------------------------------------------------------------------------
// Init / zero helpers
// ---------------------------------------------------------------------------
__global__ void gat_init_seg(unsigned* __restrict__ amax,
                             float* __restrict__ denom, int NH)
{
    int i = blockIdx.x * blockDim.x + threadIdx.x;
    if (i < NH) { amax[i] = 0u; denom[i] = 0.f; }   // 0u < enc(-inf): ordered-uint floor
}

__global__ void gat_zero(float* __restrict__ p, int n)
{
    int i = blockIdx.x * blockDim.x + threadIdx.x;
    if (i < n) p[i] = 0.f;
}

// ---------------------------------------------------------------------------
// Edge pass 1: alpha = leaky_relu(a_src[src]+a_dst[dst]); segment max via
// ordered-uint atomicMax. Edges [0,E) from edge_index, [E,E+N) = self loops.
// ---------------------------------------------------------------------------
__global__ void gat_edge_logits(const int* __restrict__ ei,
                                const float* __restrict__ a_src,
                                const float* __restrict__ a_dst,
                                float* __restrict__ alpha,
                                unsigned* __restrict__ amax,
                                int E, int N, int H)
{
    int e = blockIdx.x * blockDim.x + threadIdx.x;
    int total = E + N;
    if (e >= total) return;
    int s = (e < E) ? ei[e]     : (e - E);
    int d = (e < E) ? ei[E + e] : (e - E);
    for (int hd = 0; hd < H; ++hd) {
        float a = a_src[s * H + hd] + a_dst[d * H + hd];
        a = (a > 0.f) ? a : 0.2f * a;           // leaky_relu, slope 0.2
        alpha[(size_t)e * H + hd] = a;
        atomicMax(&amax[d * H + hd], fenc(a));
    }
}

// ---------------------------------------------------------------------------
// Edge pass 2: alpha = exp(alpha - amax[dst]); denom[dst] += alpha
// ---------------------------------------------------------------------------
__global__ void gat_edge_exp(const int* __restrict__ ei,
                             float* __restrict__ alpha,
                             const unsigned* __restrict__ amax,
                             float* __restrict__ denom,
                             int E, int N, int H)
{
    int e = blockIdx.x * blockDim.x + threadIdx.x;
    int total = E + N;
    if (e >= total) return;
    int d = (e < E) ? ei[E + e] : (e - E);
    for (int hd = 0; hd < H; ++hd) {
        float a  = alpha[(size_t)e * H + hd];
        float m  = fdec(amax[d * H + hd]);
        float ex = expf(a - m);
        alpha[(size_t)e * H + hd] = ex;
        atomicAdd(&denom[d * H + hd], ex);
    }
}

// ---------------------------------------------------------------------------
// Edge pass 3: agg[dst,h,:] += (alpha/denom[dst,h]) * H[src,h,:]
// One wave per edge; lanes cover C features (coalesced loads, f32 atomics).
// Working set fits in 192MB L2 -> atomics resolve at L2.
// ---------------------------------------------------------------------------
__global__ void gat_edge_scatter(const int* __restrict__ ei,
                                 const float* __restrict__ Hf,
                                 const float* __restrict__ alpha,
                                 const float* __restrict__ denom,
                                 float* __restrict__ agg,
                                 int E, int N, int H, int C)
{
    const int wave = (int)((blockIdx.x * blockDim.x + threadIdx.x) >> 5);
    const int lane = threadIdx.x & 31;
    const int total = E + N;
    if (wave >= total) return;
    int s = (wave < E) ? ei[wave]     : (wave - E);
    int d = (wave < E) ? ei[E + wave] : (wave - E);

    const float* hs = Hf  + (size_t)s * H * C;
    float*       od = agg + (size_t)d * H * C;
    for (int hd = 0; hd < H; ++hd) {
        float w = alpha[(size_t)wave * H + hd] / denom[d * H + hd];
        const float* hrow = hs + hd * C;
        float*       orow = od + hd * C;
        for (int c = lane; c < C; c += 32)
            atomicAdd(&orow[c], hrow[c] * w);
    }
}

// ---------------------------------------------------------------------------
// Epilogue: buf[i] = act(buf[i] + bias[i % HC])
// ---------------------------------------------------------------------------
__global__ void gat_bias_act(float* __restrict__ buf,
                             const float* __restrict__ bias,
                             int total, int HC, int doRelu)
{
    int i = blockIdx.x * blockDim.x + threadIdx.x;
    if (i >= total) return;
    float v = buf[i] + bias[i % HC];
    if (doRelu) v = fmaxf(v, 0.f);
    buf[i] = v;
}

// ---------------------------------------------------------------------------
static inline int cdiv(long long a, long long b) { return (int)((a + b - 1) / b); }

extern "C" void kernel_launch(void* const* d_in, const int* in_sizes, int n_in,
                              void* d_out, int out_size, void* d_ws, size_t ws_size,
                              hipStream_t stream)
{
    const float* x        = (const float*)d_in[0];
    const int*   ei       = (const int*)  d_in[1];   // [2,E] flat (int32)
    const float* W1       = (const float*)d_in[2];   // [256,256]
    const float* att_src1 = (const float*)d_in[3];   // [2*128]
    const float* att_dst1 = (const float*)d_in[4];
    const float* b1       = (const float*)d_in[5];   // [256]
    const float* W2       = (const float*)d_in[6];   // [128,256]
    const float* att_src2 = (const float*)d_in[7];   // [128]
    const float* att_dst2 = (const float*)d_in[8];
    const float* b2       = (const float*)d_in[9];   // [128]
    float*       out      = (float*)d_out;

    const int F  = 256;                 // IN_FEATS (layer1 K) == HEADS*HIDDEN (layer2 K)
    const int C  = 128;                 // HIDDEN
    const int H1 = 2, H2 = 1;
    const int N  = in_sizes[0] / F;     // 50000
    const int E  = in_sizes[1] / 2;     // 800000
    const int EN = E + N;

    // ---- workspace carve-up (floats) ----
    float* ws    = (float*)d_ws;
    size_t o     = 0;
    float* bufH  = ws + o; o += (size_t)N * 256;        // h features (both layers)
    float* bufX2 = ws + o; o += (size_t)N * 256;        // layer1 aggregate -> relu -> x2
    float* asrc  = ws + o; o += (size_t)N * 2;
    float* adst  = ws + o; o += (size_t)N * 2;
    unsigned* amax = (unsigned*)(ws + o); o += (size_t)N * 2;
    float* denom = ws + o; o += (size_t)N * 2;
    float* alpha = ws + o; o += (size_t)EN * 2;
    (void)ws_size; (void)n_in; (void)out_size;

    const int TB = 256;                       // 8 waves / block
    const int wavesPerBlk = TB / 32;

    // ======================= Layer 1 (H=2, concat) =======================
    {
        int tiles = cdiv(N, 16) * (256 / 16);
        gat_gemm_wmma_f32<<<cdiv(tiles, wavesPerBlk), TB, 0, stream>>>(
            x, W1, bufH, N, F, 256);

        gat_attn_dots<<<cdiv((long long)N * H1, wavesPerBlk), TB, 0, stream>>>(
            bufH, att_src1, att_dst1, asrc, adst, N, H1, C);

        gat_init_seg<<<cdiv(N * H1, TB), TB, 0, stream>>>(amax, denom, N * H1);
        gat_zero<<<cdiv((long long)N * 256, TB), TB, 0, stream>>>(bufX2, N * 256);

        gat_edge_logits<<<cdiv(EN, TB), TB, 0, stream>>>(
            ei, asrc, adst, alpha, amax, E, N, H1);
        gat_edge_exp<<<cdiv(EN, TB), TB, 0, stream>>>(
            ei, alpha, amax, denom, E, N, H1);
        gat_edge_scatter<<<cdiv(EN, wavesPerBlk), TB, 0, stream>>>(
            ei, bufH, alpha, denom, bufX2, E, N, H1, C);

        gat_bias_act<<<cdiv((long long)N * 256, TB), TB, 0, stream>>>(
            bufX2, b1, N * 256, 256, /*relu=*/1);
    }

    // ======================= Layer 2 (H=1, mean==identity) ===============
    {
        int tiles = cdiv(N, 16) * (128 / 16);
        gat_gemm_wmma_f32<<<cdiv(tiles, wavesPerBlk), TB, 0, stream>>>(
            bufX2, W2, bufH, N, F, 128);

        gat_attn_dots<<<cdiv((long long)N * H2, wavesPerBlk), TB, 0, stream>>>(
            bufH, att_src2, att_dst2, asrc, adst, N, H2, C);

        gat_init_seg<<<cdiv(N * H2, TB), TB, 0, stream>>>(amax, denom, N * H2);
        gat_zero<<<cdiv((long long)N * 128, TB), TB, 0, stream>>>(out, N * 128);

        gat_edge_logits<<<cdiv(EN, TB), TB, 0, stream>>>(
            ei, asrc, adst, alpha, amax, E, N, H2);
        gat_edge_exp<<<cdiv(EN, TB), TB, 0, stream>>>(
            ei, alpha, amax, denom, E, N, H2);
        gat_edge_scatter<<<cdiv(EN, wavesPerBlk), TB, 0, stream>>>(
            ei, bufH, alpha, denom, out, E, N, H2, C);

        gat_bias_act<<<cdiv((long long)N * 128, TB), TB, 0, stream>>>(
            out, b2, N * 128, 128, /*relu=*/0);
    }
}